// Model_54546084659574
// MI455X (gfx1250) — compile-verified
//
#include <hip/hip_runtime.h>
#include <hip/hip_bf16.h>

typedef __attribute__((ext_vector_type(2))) float v2f;
typedef __attribute__((ext_vector_type(8))) float v8f;

#define NB   128   // B*H
#define CIN  128
#define OC2  256   // 2*OUT_CH
#define WD   128
#define NG   16
#define SROW 136   // LDS row stride (pad 8 -> half-wave K-phases hit disjoint banks)

__device__ __forceinline__ v8f wmma_f32(v2f a, v2f b, v8f c) {
  // V_WMMA_F32_16X16X4_F32 : D(16x16,f32) = A(16x4,f32) * B(4x16,f32) + C
  return __builtin_amdgcn_wmma_f32_16x16x4_f32(false, a, false, b, (short)0, c,
                                               false, false);
}

// ---------------------------------------------------------------------------
// K1: qkv[nb,o,w] = sum_c W[o,c] * x[b,c,h,w]  (nb = b*64+h), + channel stats
// grid = NB*16 (o-tile per block), block = 256 (8 waves, one 16x16 w-tile each)
// ---------------------------------------------------------------------------
__global__ void k_qkv(const float* __restrict__ x, const float* __restrict__ Wq,
                      float* __restrict__ qkvBuf,
                      float* __restrict__ qkvSum, float* __restrict__ qkvSq) {
  int blk = blockIdx.x;
  int nb = blk >> 4, ot = blk & 15;
  int b = nb >> 6, h = nb & 63;
  int tid = threadIdx.x;
  int wave = tid >> 5, lane = tid & 31;
  int m = lane & 15, hi = lane >> 4, koff = hi << 1;
  int w0 = wave << 4;

  const float* Arow  = Wq + (size_t)(ot * 16 + m) * CIN;
  const float* xbase = x + (size_t)b * CIN * 64 * WD + (size_t)h * WD + w0 + m;

  v8f acc = {0.f, 0.f, 0.f, 0.f, 0.f, 0.f, 0.f, 0.f};
  for (int kb = 0; kb < CIN; kb += 4) {
    int k = kb + koff;
    v2f a;  a.x = Arow[k];            a.y = Arow[k + 1];
    const float* xp = xbase + (size_t)k * 64 * WD;
    v2f bm; bm.x = xp[0];             bm.y = xp[64 * WD];
    acc = wmma_f32(a, bm, acc);
  }

  float* obase = qkvBuf + ((size_t)nb * OC2 + ot * 16 + 8 * hi) * WD + w0 + m;
#pragma unroll
  for (int r = 0; r < 8; ++r) {
    float v = acc[r];
    obase[(size_t)r * WD] = v;
    float vs = v, vq = v * v;
    vs += __shfl_xor(vs, 1); vq += __shfl_xor(vq, 1);
    vs += __shfl_xor(vs, 2); vq += __shfl_xor(vq, 2);
    vs += __shfl_xor(vs, 4); vq += __shfl_xor(vq, 4);
    vs += __shfl_xor(vs, 8); vq += __shfl_xor(vq, 8);
    if (m == 0) {
      int o = ot * 16 + r + 8 * hi;
      atomicAdd(&qkvSum[o], vs);
      atomicAdd(&qkvSq[o], vq);
    }
  }
}

// ---------------------------------------------------------------------------
// BN finalize: scale = gamma*rsqrt(var+eps), bias = beta - mean*scale
// ---------------------------------------------------------------------------
__global__ void k_finalize(const float* __restrict__ sum, const float* __restrict__ sq,
                           const float* __restrict__ gamma, const float* __restrict__ beta,
                           float* __restrict__ scale, float* __restrict__ bias,
                           int nch, float invN) {
  int i = blockIdx.x * blockDim.x + threadIdx.x;
  if (i < nch) {
    float mu = sum[i] * invN;
    float va = sq[i] * invN - mu * mu;
    float s  = gamma[i] * rsqrtf(va + 1e-5f);
    scale[i] = s;
    bias[i]  = beta[i] - mu * s;
  }
}

// ---------------------------------------------------------------------------
// Shared helper: build QE[t][i], KE[t][i] in LDS (stride SROW).
// QE[t][i] = sum_{c<4} qn[c,t] * rel_emb[c, t-i+127], KE uses rows 4..7.
// Returns per-thread partial sums if wanted.
// ---------------------------------------------------------------------------
__device__ __forceinline__ void build_qeke(float* QE, float* KE,
                                           const float* __restrict__ qkvBuf,
                                           const float* __restrict__ relEmb,
                                           const float* __restrict__ qs,
                                           const float* __restrict__ qb,
                                           int nb, int g, int tid,
                                           float* qeS, float* qeQ,
                                           float* keS, float* keQ) {
  int t  = tid >> 1;
  int i0 = (tid & 1) << 6;
  const float* grp = qkvBuf + ((size_t)nb * OC2 + g * 16) * WD;
  float qn[4], kn[4];
#pragma unroll
  for (int c = 0; c < 4; ++c) {
    qn[c] = grp[c * WD + t]       * qs[g * 16 + c]     + qb[g * 16 + c];
    kn[c] = grp[(4 + c) * WD + t] * qs[g * 16 + 4 + c] + qb[g * 16 + 4 + c];
  }
  float s0 = 0, q0 = 0, s1 = 0, q1 = 0;
  for (int ii = 0; ii < 64; ++ii) {
    int i = i0 + ii;
    int e = t - i + 127;  // in [0,254]
    float aq = 0, ak = 0;
#pragma unroll
    for (int c = 0; c < 4; ++c) {
      aq += qn[c] * relEmb[c * 255 + e];
      ak += kn[c] * relEmb[(4 + c) * 255 + e];
    }
    QE[t * SROW + i] = aq; s0 += aq; q0 += aq * aq;
    KE[t * SROW + i] = ak; s1 += ak; q1 += ak * ak;
  }
  *qeS = s0; *qeQ = q0; *keS = s1; *keQ = q1;
}

// ---------------------------------------------------------------------------
// K3: sim BN statistics. One block per (nb,g). qk via WMMA, nothing stored.
// ---------------------------------------------------------------------------
__global__ void k_simstats(const float* __restrict__ qkvBuf, const float* __restrict__ relEmb,
                           const float* __restrict__ qs, const float* __restrict__ qb,
                           float* __restrict__ simSum, float* __restrict__ simSq) {
  extern __shared__ float sm[];
  float* QE = sm;
  float* KE = sm + 128 * SROW;
  __shared__ float blkstat[6];
  int blk = blockIdx.x;
  int nb = blk >> 4, g = blk & 15;
  int tid = threadIdx.x;
  if (tid < 6) blkstat[tid] = 0.f;
  __syncthreads();

  float qeS, qeQ, keS, keQ;
  build_qeke(QE, KE, qkvBuf, relEmb, qs, qb, nb, g, tid, &qeS, &qeQ, &keS, &keQ);
#pragma unroll
  for (int d = 1; d < 32; d <<= 1) {
    qeS += __shfl_xor(qeS, d); qeQ += __shfl_xor(qeQ, d);
    keS += __shfl_xor(keS, d); keQ += __shfl_xor(keQ, d);
  }
  if ((tid & 31) == 0) {
    atomicAdd(&blkstat[0], qeS); atomicAdd(&blkstat[1], qeQ);
    atomicAdd(&blkstat[2], keS); atomicAdd(&blkstat[3], keQ);
  }
  __syncthreads();

  int wave = tid >> 5, lane = tid & 31;
  int m = lane & 15, hi = lane >> 4, koff = hi << 1;
  int i0g = wave << 4;
  float qkS = 0, qkQ = 0;
  for (int jt = 0; jt < 8; ++jt) {
    int j0 = jt << 4;
    v8f acc = {0.f, 0.f, 0.f, 0.f, 0.f, 0.f, 0.f, 0.f};
    for (int kb = 0; kb < 128; kb += 4) {
      int k = kb + koff;
      v2f a;  a.x = QE[k * SROW + i0g + m];  a.y = QE[(k + 1) * SROW + i0g + m];
      v2f bm; bm.x = KE[k * SROW + j0 + m];  bm.y = KE[(k + 1) * SROW + j0 + m];
      acc = wmma_f32(a, bm, acc);
    }
#pragma unroll
    for (int r = 0; r < 8; ++r) { float v = acc[r]; qkS += v; qkQ += v * v; }
  }
#pragma unroll
  for (int d = 1; d < 32; d <<= 1) { qkS += __shfl_xor(qkS, d); qkQ += __shfl_xor(qkQ, d); }
  if (lane == 0) { atomicAdd(&blkstat[4], qkS); atomicAdd(&blkstat[5], qkQ); }
  __syncthreads();
  if (tid == 0) {
    atomicAdd(&simSum[g],      blkstat[4]); atomicAdd(&simSq[g],      blkstat[5]);
    atomicAdd(&simSum[16 + g], blkstat[0]); atomicAdd(&simSq[16 + g], blkstat[1]);
    atomicAdd(&simSum[32 + g], blkstat[2]); atomicAdd(&simSq[32 + g], blkstat[3]);
  }
}

// ---------------------------------------------------------------------------
// K5: rebuild qe/ke, WMMA qk, fused BN-sum + softmax (wave-local rows),
// then am/ame + out-BN stats. One block per (nb,g).
// ---------------------------------------------------------------------------
__global__ void k_apply(const float* __restrict__ qkvBuf, const float* __restrict__ relEmb,
                        const float* __restrict__ qs, const float* __restrict__ qb,
                        const float* __restrict__ ss, const float* __restrict__ sb,
                        float* __restrict__ outBuf,
                        float* __restrict__ outSum, float* __restrict__ outSq) {
  extern __shared__ float sm[];
  float* QE = sm;                 // reused as S (softmax probs) later
  float* KE = sm + 128 * SROW;
  __shared__ float vnLDS[8 * 128];
  int blk = blockIdx.x;
  int nb = blk >> 4, g = blk & 15;
  int tid = threadIdx.x;

  float d0, d1, d2, d3;
  build_qeke(QE, KE, qkvBuf, relEmb, qs, qb, nb, g, tid, &d0, &d1, &d2, &d3);

  const float* grp = qkvBuf + ((size_t)nb * OC2 + g * 16) * WD;
  for (int e = tid; e < 1024; e += 256) {
    int c = e >> 7, j = e & 127;
    int ch = g * 16 + 8 + c;
    vnLDS[e] = grp[(8 + c) * WD + j] * qs[ch] + qb[ch];
  }
  __syncthreads();

  int wave = tid >> 5, lane = tid & 31;
  int m = lane & 15, hi = lane >> 4, koff = hi << 1;
  int i0g = wave << 4;  // wave owns rows [i0g, i0g+16) -> full softmax rows
  float scQK = ss[g], scQE = ss[16 + g], scKE = ss[32 + g];
  float bAll = sb[g] + sb[16 + g] + sb[32 + g];

  v8f acc[8];
  for (int jt = 0; jt < 8; ++jt) {
    int j0 = jt << 4;
    v8f a8 = {0.f, 0.f, 0.f, 0.f, 0.f, 0.f, 0.f, 0.f};
    for (int kb = 0; kb < 128; kb += 4) {
      int k = kb + koff;
      v2f a;  a.x = QE[k * SROW + i0g + m];  a.y = QE[(k + 1) * SROW + i0g + m];
      v2f bm; bm.x = KE[k * SROW + j0 + m];  bm.y = KE[(k + 1) * SROW + j0 + m];
      a8 = wmma_f32(a, bm, a8);
    }
    acc[jt] = a8;
  }

  // sim = bn(qk)+bn(qe)+bn(ke)  (pointwise QE/KE reads BEFORE any overwrite)
#pragma unroll
  for (int jt = 0; jt < 8; ++jt) {
    int j = (jt << 4) + m;
#pragma unroll
    for (int r = 0; r < 8; ++r) {
      int i = i0g + r + 8 * hi;
      acc[jt][r] = acc[jt][r] * scQK + QE[i * SROW + j] * scQE +
                   KE[i * SROW + j] * scKE + bAll;
    }
  }
  // softmax over j: row i lives in one 16-lane half across 8 j-tiles
#pragma unroll
  for (int r = 0; r < 8; ++r) {
    float mx = acc[0][r];
#pragma unroll
    for (int jt = 1; jt < 8; ++jt) mx = fmaxf(mx, acc[jt][r]);
    mx = fmaxf(mx, __shfl_xor(mx, 1));
    mx = fmaxf(mx, __shfl_xor(mx, 2));
    mx = fmaxf(mx, __shfl_xor(mx, 4));
    mx = fmaxf(mx, __shfl_xor(mx, 8));
    float sum = 0.f;
#pragma unroll
    for (int jt = 0; jt < 8; ++jt) {
      float ex = __expf(acc[jt][r] - mx);
      acc[jt][r] = ex;
      sum += ex;
    }
    sum += __shfl_xor(sum, 1);
    sum += __shfl_xor(sum, 2);
    sum += __shfl_xor(sum, 4);
    sum += __shfl_xor(sum, 8);
    float inv = 1.f / sum;
#pragma unroll
    for (int jt = 0; jt < 8; ++jt) acc[jt][r] *= inv;
  }
  __syncthreads();  // all reads of QE/KE complete before overwriting QE with S
#pragma unroll
  for (int jt = 0; jt < 8; ++jt) {
    int j = (jt << 4) + m;
#pragma unroll
    for (int r = 0; r < 8; ++r) {
      int i = i0g + r + 8 * hi;
      QE[i * SROW + j] = acc[jt][r];
    }
  }
  __syncthreads();

  // am[c,i] = sum_j S[i,j]*vn[c,j]; ame[c,i] = sum_j S[i,j]*rel_emb[8+c,i-j+127]
  for (int kk = 0; kk < 4; ++kk) {
    int id = tid + (kk << 8);
    int c = id >> 7, i = id & 127;     // c uniform within each wave
    const float* Srow = QE + i * SROW;
    const float* ve = relEmb + (8 + c) * 255 + i + 127;
    float am = 0.f, ame = 0.f;
    for (int j = 0; j < 128; ++j) {
      float s = Srow[j];
      am  += s * vnLDS[(c << 7) + j];
      ame += s * ve[-j];
    }
    int o0 = g * 16 + (c << 1);
    outBuf[((size_t)nb * OC2 + o0) * WD + i]     = am;
    outBuf[((size_t)nb * OC2 + o0 + 1) * WD + i] = ame;

    float s0 = am, q0 = am * am, s1 = ame, q1 = ame * ame;
#pragma unroll
    for (int d = 1; d < 32; d <<= 1) {
      s0 += __shfl_xor(s0, d); q0 += __shfl_xor(q0, d);
      s1 += __shfl_xor(s1, d); q1 += __shfl_xor(q1, d);
    }
    if (lane == 0) {
      atomicAdd(&outSum[o0], s0);     atomicAdd(&outSq[o0], q0);
      atomicAdd(&outSum[o0 + 1], s1); atomicAdd(&outSq[o0 + 1], q1);
    }
  }
}

// ---------------------------------------------------------------------------
// K7: y[b,oc,h,w] = bn(out)[nb,2oc,w] + bn(out)[nb,2oc+1,w]
// ---------------------------------------------------------------------------
__global__ void k_final(const float* __restrict__ outBuf,
                        const float* __restrict__ os, const float* __restrict__ ob,
                        float* __restrict__ y) {
  int idx = blockIdx.x * 256 + threadIdx.x;
  if (idx >= 2 * 128 * 64 * 128) return;
  int w  = idx & 127;
  int h  = (idx >> 7) & 63;
  int oc = (idx >> 13) & 127;
  int b  = idx >> 20;
  int nb = b * 64 + h;
  int o0 = oc * 2;
  float v0 = outBuf[((size_t)nb * OC2 + o0) * WD + w];
  float v1 = outBuf[((size_t)nb * OC2 + o0 + 1) * WD + w];
  y[idx] = (v0 * os[o0] + ob[o0]) + (v1 * os[o0 + 1] + ob[o0 + 1]);
}

// ---------------------------------------------------------------------------
extern "C" void kernel_launch(void* const* d_in, const int* in_sizes, int n_in,
                              void* d_out, int out_size, void* d_ws, size_t ws_size,
                              hipStream_t stream) {
  const float* x   = (const float*)d_in[0];
  const float* Wq  = (const float*)d_in[1];
  const float* gq  = (const float*)d_in[2];
  const float* bq  = (const float*)d_in[3];
  const float* rel = (const float*)d_in[4];
  const float* gs  = (const float*)d_in[5];
  const float* bs  = (const float*)d_in[6];
  const float* go  = (const float*)d_in[7];
  const float* bo  = (const float*)d_in[8];
  float* y = (float*)d_out;

  float* ws = (float*)d_ws;
  float* qkvBuf = ws;                               // 128*256*128 = 4194304 f
  float* outBuf = qkvBuf + 4194304;                 // 4194304 f
  float* stats  = outBuf + 4194304;                 // 1120 f
  float* qkvSum = stats;        float* qkvSq = stats + 256;
  float* simSum = stats + 512;  float* simSq = stats + 560;
  float* outSum = stats + 608;  float* outSq = stats + 864;
  float* par    = stats + 1120;                     // 1120 f
  float* qkvSc = par;        float* qkvBi = par + 256;
  float* simSc = par + 512;  float* simBi = par + 560;
  float* outSc = par + 608;  float* outBi = par + 864;

  hipMemsetAsync(stats, 0, 1120 * sizeof(float), stream);

  size_t lds = (size_t)2 * 128 * SROW * sizeof(float);  // 139264 B

  k_qkv<<<NB * 16, 256, 0, stream>>>(x, Wq, qkvBuf, qkvSum, qkvSq);
  k_finalize<<<1, 256, 0, stream>>>(qkvSum, qkvSq, gq, bq, qkvSc, qkvBi,
                                    256, 1.f / 16384.f);
  k_simstats<<<NB * NG, 256, lds, stream>>>(qkvBuf, rel, qkvSc, qkvBi,
                                            simSum, simSq);
  k_finalize<<<1, 64, 0, stream>>>(simSum, simSq, gs, bs, simSc, simBi,
                                   48, 1.f / 2097152.f);
  k_apply<<<NB * NG, 256, lds, stream>>>(qkvBuf, rel, qkvSc, qkvBi,
                                         simSc, simBi, outBuf, outSum, outSq);
  k_finalize<<<1, 256, 0, stream>>>(outSum, outSq, go, bo, outSc, outBi,
                                    256, 1.f / 16384.f);
  k_final<<<(2097152 + 255) / 256, 256, 0, stream>>>(outBuf, outSc, outBi, y);
}